// RealS5ZOHLayer_61031485276179
// MI455X (gfx1250) — compile-verified
//
#include <hip/hip_runtime.h>
#include <hip/hip_bf16.h>
#include <math.h>

// ---------------- problem constants ----------------
#define BB   4      // batch
#define TT   4096   // time steps
#define DM   1024   // d_model
#define GG   8      // groups
#define GD   128    // group dim
#define NST  32     // state dim
#define LCH  64     // chunk length (timesteps per block)
#define NCH  (TT / LCH)   // 64 chunks
#define SEG  8      // per-thread scan segment length
#define NSEG (LCH / SEG)  // 8 segments

// padded LDS row strides (bank-conflict-free for the WMMA fragment patterns)
#define US   132    // 128-wide row stride (floats): 132 mod 64 = 4 -> conflict-free frag access
#define BS   36     // 32-wide row stride (floats): conflict-free frag access, 16B aligned rows

typedef float v2f __attribute__((ext_vector_type(2)));
typedef float v8f __attribute__((ext_vector_type(8)));

// D = A(16x4 f32) x B(4x16 f32) + C(16x16 f32), full fp32 matrix op on CDNA5.
__device__ __forceinline__ v8f wmma4(v2f a, v2f b, v8f c) {
  return __builtin_amdgcn_wmma_f32_16x16x4_f32(
      /*neg_a=*/false, a, /*neg_b=*/false, b,
      /*c_mod=*/(short)0, c, /*reuse_a=*/false, /*reuse_b=*/false);
}

// ---------------- kernel 0: parameter precompute ----------------
// tid -> (g,n). Computes Abar, Abar^8, Abar^64, Bbar = frac * B_param.
__global__ __launch_bounds__(256) void k0_setup(
    const float* __restrict__ A_log, const float* __restrict__ B_param,
    const float* __restrict__ dt_logit,
    float* __restrict__ wAbar, float* __restrict__ wAbar8,
    float* __restrict__ wAbar64, float* __restrict__ wBbar) {
  int tid = threadIdx.x;          // 0..255
  int g = tid >> 5;
  float x = A_log[tid];
  float sp = (x > 20.f) ? x : log1pf(expf(x));   // softplus
  float A = -sp;
  float dtl = dt_logit[g];
  float sig = 1.f / (1.f + expf(-dtl));
  float dt = 0.001f + (0.1f - 0.001f) * sig;
  float Abar = expf(A * dt);
  float frac = (fabsf(A) > 1e-6f) ? (Abar - 1.f) / A : dt;
  wAbar[tid] = Abar;
  float p = Abar;
  p *= p; p *= p; p *= p;          // Abar^8
  wAbar8[tid] = p;
  p *= p; p *= p; p *= p;          // Abar^64
  wAbar64[tid] = p;
  const float* bp = B_param + (size_t)tid * GD;
  float*       bb = wBbar   + (size_t)tid * GD;
#pragma unroll 4
  for (int i = 0; i < GD; ++i) bb[i] = frac * bp[i];
}

// ---------------- kernel 1: per-chunk Bu GEMM + local (zero-init) scan ----------------
// grid = BB*GG*NCH blocks, 256 threads (8 waves).
__global__ __launch_bounds__(256) void k1_bu_scan(
    const float* __restrict__ u,
    const float* __restrict__ wAbar, const float* __restrict__ wAbar8,
    const float* __restrict__ wBbar, float* __restrict__ wLocal) {
  __shared__ float lds_u[LCH * US];      // 33 KB: u chunk [t][i], padded
  __shared__ float lds_bu[LCH * BS];     //  9 KB: Bu -> local states [t][n], padded
  __shared__ float segsum[NSEG * NST];
  __shared__ float segcar[NSEG * NST];

  int blk = blockIdx.x;
  int c = blk & (NCH - 1);
  int g = (blk >> 6) & (GG - 1);
  int b = blk >> 9;
  int tid = threadIdx.x;

  // warm L0/L2 with this group's Bbar panel (16 KB) while we stage u
  const float* Bb = wBbar + (size_t)g * NST * GD;
  __builtin_prefetch(Bb + tid * 16, 0, 3);   // 256 threads x 64B = whole panel

  // --- stage u chunk (64 rows x 128 floats), fully coalesced float4 ---
  const float* ub = u + ((size_t)(b * TT + c * LCH)) * DM + g * GD;
#pragma unroll
  for (int j = 0; j < 8; ++j) {
    int idx = tid + 256 * j;             // float4 index, 0..2047
    int tl = idx >> 5;                   // row 0..63
    int q  = idx & 31;                   // float4 col 0..31
    float4 v = *(const float4*)(ub + (size_t)tl * DM + q * 4);
    *(float4*)(&lds_u[tl * US + q * 4]) = v;
  }
  __syncthreads();

  // --- Bu = u_chunk(64x128) x Bbar^T(128x32), fp32 WMMA 16x16x4 ---
  int lane   = tid & 31;
  int w      = tid >> 5;                 // wave 0..7
  int ttile  = w >> 1;                   // 4 t-tiles of 16 rows
  int ntile  = w & 1;                    // 2 n-tiles of 16 cols
  int m      = lane & 15;
  int koff   = (lane >> 4) << 1;         // 0 or 2
  int rowadd = (lane >> 4) << 3;         // +8 for upper half lanes (D layout)

  const float* arow = &lds_u[(ttile * 16 + m) * US];
  const float* brow = Bb + (ntile * 16 + m) * GD;   // B[k][n] = Bbar[n][k]
  v8f acc = {};
#pragma unroll
  for (int k = 0; k < GD; k += 4) {
    v2f a  = *(const v2f*)(arow + k + koff);
    v2f bf = *(const v2f*)(brow + k + koff);
    acc = wmma4(a, bf, acc);
  }
#pragma unroll
  for (int r = 0; r < 8; ++r) {
    int t = ttile * 16 + r + rowadd;
    lds_bu[t * BS + ntile * 16 + m] = acc[r];
  }
  __syncthreads();

  // --- hierarchical within-chunk scan: s_t = Abar*s_{t-1} + Bu_t, zero init ---
  int n = tid & 31;
  int s = tid >> 5;                      // segment 0..7 (== wave id)
  float Ab = wAbar[g * NST + n];
  float a8 = wAbar8[g * NST + n];
  float run = 0.f;
#pragma unroll
  for (int k = 0; k < SEG; ++k) {
    int t = s * SEG + k;
    run = run * Ab + lds_bu[t * BS + n];
    lds_bu[t * BS + n] = run;
  }
  segsum[s * NST + n] = run;
  __syncthreads();
  if (s == 0) {                          // per-n serial carry over 8 segments
    float carry = 0.f;
#pragma unroll
    for (int ss = 0; ss < NSEG; ++ss) {
      segcar[ss * NST + n] = carry;
      carry = carry * a8 + segsum[ss * NST + n];
    }
  }
  __syncthreads();
  {
    float carry = segcar[s * NST + n];
    float f = Ab;
#pragma unroll
    for (int k = 0; k < SEG; ++k) {
      lds_bu[(s * SEG + k) * BS + n] += f * carry;
      f *= Ab;
    }
  }
  __syncthreads();

  // --- store local states (zero-init at chunk start) to workspace ---
  float* out = wLocal + ((size_t)((b * GG + g) * TT + c * LCH)) * NST;
#pragma unroll
  for (int j = 0; j < 2; ++j) {
    int idx4 = tid + 256 * j;            // float4 index, 0..511
    int tl = idx4 >> 3;
    int q  = idx4 & 7;
    *(float4*)(out + tl * NST + q * 4) = *(const float4*)(&lds_bu[tl * BS + q * 4]);
  }
}

// ---------------- kernel 2: inter-chunk carry recurrence + final state ----------------
__global__ __launch_bounds__(256) void k2_carry(
    const float* __restrict__ state0, const float* __restrict__ wAbar64,
    const float* __restrict__ wLocal, float* __restrict__ wCarry,
    float* __restrict__ sfinal) {
  int tg = blockIdx.x * 256 + threadIdx.x;  // 0..1023 -> (b,g,n)
  int n = tg & 31;
  int g = (tg >> 5) & 7;
  int b = tg >> 8;
  int bg = b * GG + g;
  float carry = state0[bg * NST + n];
  float aL = wAbar64[g * NST + n];
  const float* loc = wLocal + (size_t)bg * TT * NST;
  for (int cc = 0; cc < NCH; ++cc) {
    wCarry[(bg * NCH + cc) * NST + n] = carry;                 // state before chunk cc
    carry = carry * aL + loc[((size_t)cc * LCH + (LCH - 1)) * NST + n];
  }
  sfinal[bg * NST + n] = carry;
}

// ---------------- kernel 3: apply carry + Cs GEMM + y (coalesced epilogue) ----------------
__global__ __launch_bounds__(256) void k3_y(
    const float* __restrict__ u, const float* __restrict__ Cmat,
    const float* __restrict__ Dp,
    const float* __restrict__ wAbar, const float* __restrict__ wAbar8,
    const float* __restrict__ wLocal, const float* __restrict__ wCarry,
    float* __restrict__ y) {
  __shared__ float lds_s[LCH * BS];      //  9 KB: states [t][n], padded
  __shared__ float lds_y[LCH * US];      // 33 KB: Cs tile [t][d] staging, padded

  int blk = blockIdx.x;
  int c = blk & (NCH - 1);
  int g = (blk >> 6) & (GG - 1);
  int b = blk >> 9;
  int bg = b * GG + g;
  int tid = threadIdx.x;

  // warm L0/L2 with this group's C panel (16 KB) while we stage states
  const float* Cg = Cmat + (size_t)g * GD * NST;
  __builtin_prefetch(Cg + tid * 16, 0, 3);   // 256 threads x 64B = whole panel

  const float* src = wLocal + ((size_t)(bg * TT + c * LCH)) * NST;
#pragma unroll
  for (int j = 0; j < 2; ++j) {
    int idx4 = tid + 256 * j;
    int tl = idx4 >> 3;
    int q  = idx4 & 7;
    *(float4*)(&lds_s[tl * BS + q * 4]) = *(const float4*)(src + tl * NST + q * 4);
  }
  __syncthreads();

  {   // states[t] = local[t] + Abar^{t+1} * carry_chunk
    int n = tid & 31;
    int s = tid >> 5;
    float carry = wCarry[(bg * NCH + c) * NST + n];
    float Ab = wAbar[g * NST + n];
    float a8 = wAbar8[g * NST + n];
    float f = Ab;
    for (int j = 0; j < s; ++j) f *= a8;   // Abar^(s*SEG+1)
#pragma unroll
    for (int k = 0; k < SEG; ++k) {
      lds_s[(s * SEG + k) * BS + n] += f * carry;
      f *= Ab;
    }
  }
  __syncthreads();

  // --- Cs = states(64x32) x C^T(32x128), fp32 WMMA; stage tile in LDS ---
  int lane   = tid & 31;
  int w      = tid >> 5;
  int ttile  = w >> 1;                   // 4 t-tiles
  int dhalf  = w & 1;                    // d cols [0,64) or [64,128)
  int m      = lane & 15;
  int koff   = (lane >> 4) << 1;
  int rowadd = (lane >> 4) << 3;

  v2f afrag[8];                          // A fragments, k = 0..31 step 4
  const float* arow = &lds_s[(ttile * 16 + m) * BS];
#pragma unroll
  for (int k8 = 0; k8 < 8; ++k8)
    afrag[k8] = *(const v2f*)(arow + k8 * 4 + koff);

#pragma unroll
  for (int dt = 0; dt < 4; ++dt) {
    int d = dhalf * 64 + dt * 16 + m;    // output column (per lane)
    const float* brow = Cg + (size_t)d * NST;   // B[k][d] = C[g][d][k]
    v8f acc = {};
#pragma unroll
    for (int k8 = 0; k8 < 8; ++k8) {
      v2f bf = *(const v2f*)(brow + k8 * 4 + koff);
      acc = wmma4(afrag[k8], bf, acc);
    }
#pragma unroll
    for (int r = 0; r < 8; ++r) {        // rows t and t+8 hit disjoint bank ranges (US=132)
      int t = ttile * 16 + r + rowadd;
      lds_y[t * US + d] = acc[r];
    }
  }
  __syncthreads();

  // --- fused epilogue: y = Cs + Dp*u, fully coalesced float4 traffic ---
  const float* ubase = u + ((size_t)(b * TT + c * LCH)) * DM + g * GD;
  float*       ybase = y + ((size_t)(b * TT + c * LCH)) * DM + g * GD;
  const float* dpg   = Dp + g * GD;
#pragma unroll
  for (int j = 0; j < 8; ++j) {
    int idx = tid + 256 * j;             // float4 index, 0..2047
    int tl = idx >> 5;                   // row 0..63
    int q  = idx & 31;                   // float4 col 0..31
    float4 uv = *(const float4*)(ubase + (size_t)tl * DM + q * 4);
    float4 cs = *(const float4*)(&lds_y[tl * US + q * 4]);
    float4 dp = *(const float4*)(dpg + q * 4);
    float4 o;
    o.x = cs.x + dp.x * uv.x;
    o.y = cs.y + dp.y * uv.y;
    o.z = cs.z + dp.z * uv.z;
    o.w = cs.w + dp.w * uv.w;
    *(float4*)(ybase + (size_t)tl * DM + q * 4) = o;
  }
}

// ---------------- launcher ----------------
extern "C" void kernel_launch(void* const* d_in, const int* in_sizes, int n_in,
                              void* d_out, int out_size, void* d_ws, size_t ws_size,
                              hipStream_t stream) {
  (void)in_sizes; (void)n_in; (void)out_size; (void)ws_size;
  const float* u        = (const float*)d_in[0];
  const float* state0   = (const float*)d_in[1];
  const float* A_log    = (const float*)d_in[2];
  const float* B_param  = (const float*)d_in[3];
  const float* Cmat     = (const float*)d_in[4];
  const float* Dp       = (const float*)d_in[5];
  const float* dt_logit = (const float*)d_in[6];
  float* out = (float*)d_out;
  float* ws  = (float*)d_ws;

  // workspace layout (floats)
  float* wAbar   = ws;                    // 256
  float* wAbar8  = ws + 256;              // 256
  float* wAbar64 = ws + 512;              // 256
  float* wBbar   = ws + 768;              // 8*32*128 = 32768
  float* wLocal  = ws + 33536;            // 4*8*4096*32 = 4,194,304 (16 MB)
  float* wCarry  = wLocal + (size_t)BB * GG * TT * NST;  // 4*8*64*32 = 65536

  float* y      = out;
  float* sfinal = out + (size_t)BB * TT * DM;

  k0_setup<<<1, 256, 0, stream>>>(A_log, B_param, dt_logit,
                                  wAbar, wAbar8, wAbar64, wBbar);
  k1_bu_scan<<<BB * GG * NCH, 256, 0, stream>>>(u, wAbar, wAbar8, wBbar, wLocal);
  k2_carry<<<4, 256, 0, stream>>>(state0, wAbar64, wLocal, wCarry, sfinal);
  k3_y<<<BB * GG * NCH, 256, 0, stream>>>(u, Cmat, Dp, wAbar, wAbar8,
                                          wLocal, wCarry, y);
}